// GroupedQueryAttention_89163521064963
// MI455X (gfx1250) — compile-verified
//
#include <hip/hip_runtime.h>

typedef __attribute__((ext_vector_type(2))) float v2f;
typedef __attribute__((ext_vector_type(8))) float v8f;

#define BDIM 2
#define TDIM 2048
#define EDIM 2048
#define HH   32
#define KVH  8
#define DH   64
#define RR   (HH / KVH)
#define CT   8            // column tiles per wave in GEMM (8*16 = 128 cols)

__device__ __forceinline__ v8f wmma4(v2f a, v2f b, v8f c) {
  // D = A(16x4,f32) * B(4x16,f32) + C(16x16,f32)  -> v_wmma_f32_16x16x4_f32
  return __builtin_amdgcn_wmma_f32_16x16x4_f32(
      /*neg_a=*/false, a, /*neg_b=*/false, b,
      /*c_mod=*/(short)0, c, /*reuse_a=*/false, /*reuse_b=*/false);
}

// ---------------------------------------------------------------------------
// fp32 WMMA GEMM: C[M,N] = A[M,K] @ W[K,N].
// One wave per 16x128 output strip: the A fragment is loaded once per k-step
// and reused across 8 WMMAs (8 accumulators), cutting A traffic 8x and the
// vmem:wmma issue ratio to ~2:1.
// ---------------------------------------------------------------------------
__global__ void __launch_bounds__(256)
gemm16x128_f32(const float* __restrict__ A, const float* __restrict__ W,
               float* __restrict__ C, int M, int K, int N) {
  const int lane  = threadIdx.x & 31;
  const int wave  = blockIdx.x * (blockDim.x >> 5) + (threadIdx.x >> 5);
  const int tilesN = N / (16 * CT);
  const int tm = wave / tilesN;
  const int tn = wave - tm * tilesN;
  if (tm * 16 >= M) return;               // wave-uniform exit (EXEC stays full)

  const int half = lane >> 4;             // selects K pair {0,1} vs {2,3}
  const int l16  = lane & 15;
  const int arow = tm * 16 + l16;
  const int col0 = tn * (16 * CT) + l16;

  const float* __restrict__ ap = A + (size_t)arow * K + half * 2;

  v8f acc[CT];
  v8f zero = {};
#pragma unroll
  for (int c = 0; c < CT; ++c) acc[c] = zero;

#pragma unroll 2
  for (int k0 = 0; k0 < K; k0 += 4) {
    v2f a = *(const v2f*)(ap + k0);                          // A[arow][k0+2h..+1]
    const float* wp = W + (size_t)(k0 + half * 2) * N + col0;
#pragma unroll
    for (int c = 0; c < CT; ++c) {
      v2f b; b.x = wp[c * 16]; b.y = wp[N + c * 16];         // W[k][col], W[k+1][col]
      acc[c] = wmma4(a, b, acc[c]);
    }
  }

  float* cp = C + (size_t)(tm * 16 + half * 8) * N + col0;   // VGPR r -> row r / r+8
#pragma unroll
  for (int c = 0; c < CT; ++c) {
#pragma unroll
    for (int r = 0; r < 8; ++r) cp[(size_t)r * N + c * 16] = acc[c][r];
  }
}

// ---------------------------------------------------------------------------
// RoPE, in place on [B,T,heads,DH]; pair i is contiguous (2i, 2i+1).
// ---------------------------------------------------------------------------
__global__ void rope_apply(float* __restrict__ x, const float* __restrict__ fc,
                           int heads, int total_pairs) {
  int i = blockIdx.x * blockDim.x + threadIdx.x;
  if (i >= total_pairs) return;
  const int hd = DH / 2;
  int di  = i % hd;
  int rem = i / hd;                 // (b*T + t)*heads + h
  int t   = (rem / heads) % TDIM;
  float c = fc[(t * hd + di) * 2 + 0];
  float s = fc[(t * hd + di) * 2 + 1];
  float* p = x + (size_t)i * 2;
  float a = p[0], b = p[1];
  p[0] = a * c - b * s;
  p[1] = a * s + b * c;
}

// ---------------------------------------------------------------------------
// Causal GQA flash attention. One wave (one 32-thread block) per
// (batch, head, 16-row query tile). All WMMA in fp32.
// ---------------------------------------------------------------------------
__global__ void __launch_bounds__(32)
gqa_attention(const float* __restrict__ Q, const float* __restrict__ K,
              const float* __restrict__ V, float* __restrict__ O) {
  __shared__ float pbuf[256];                       // 16x16 P tile staging
  const int lane = threadIdx.x;
  const int half = lane >> 4;
  const int l16  = lane & 15;

  int id = blockIdx.x;
  const int qt = id % (TDIM / 16);  id /= (TDIM / 16);
  const int h  = id % HH;           const int b = id / HH;
  const int kvh   = h / RR;
  const int qbase = qt * 16;
  const float scale = 0.125f;                       // 1/sqrt(64)

  // Q tile resident in A-layout registers, pre-scaled.
  v2f qa[16];
  const float* qrow = Q + ((size_t)(b * TDIM + qbase + l16) * HH + h) * DH + half * 2;
#pragma unroll
  for (int kk = 0; kk < 16; ++kk) {
    v2f t = *(const v2f*)(qrow + kk * 4);
    qa[kk].x = t.x * scale;  qa[kk].y = t.y * scale;
  }

  v8f o0 = {}, o1 = {}, o2 = {}, o3 = {};
  float m[8], l[8];
#pragma unroll
  for (int r = 0; r < 8; ++r) { m[r] = -__builtin_inff(); l[r] = 0.0f; }

  for (int kt = 0; kt <= qt; ++kt) {
    const int kb0 = kt * 16;

    // S = (Q*scale) @ K^T   (16 steps of 16x16x4)
    v8f s = {};
    const float* kcol = K + ((size_t)(b * TDIM + kb0 + l16) * KVH + kvh) * DH + half * 2;
#pragma unroll
    for (int kk = 0; kk < 16; ++kk) {
      v2f kb = *(const v2f*)(kcol + kk * 4);
      s = wmma4(qa[kk], kb, s);
    }

    // Causal mask on the diagonal tile.
    if (kt == qt) {
#pragma unroll
      for (int r = 0; r < 8; ++r) {
        int row = r + half * 8;
        if (l16 > row) s[r] = -__builtin_inff();
      }
    }

    // Online softmax: per-row max/sum via shfl_xor within the 16-lane half.
#pragma unroll
    for (int r = 0; r < 8; ++r) {
      float mx = s[r];
#pragma unroll
      for (int sh = 1; sh < 16; sh <<= 1) mx = fmaxf(mx, __shfl_xor(mx, sh, 32));
      float mn    = fmaxf(m[r], mx);
      float alpha = __expf(m[r] - mn);
      float p     = __expf(s[r] - mn);
      float ps    = p;
#pragma unroll
      for (int sh = 1; sh < 16; sh <<= 1) ps += __shfl_xor(ps, sh, 32);
      l[r] = l[r] * alpha + ps;
      m[r] = mn;
      o0[r] *= alpha;  o1[r] *= alpha;  o2[r] *= alpha;  o3[r] *= alpha;
      pbuf[(r + half * 8) * 16 + l16] = p;          // C-layout -> LDS
    }
    asm volatile("s_wait_dscnt 0" ::: "memory");    // drain LDS stores (one wave)

    // O += P @ V   (K-dim 16 -> 4 steps; N=64 -> 4 output tiles)
    const float* vrow = V + ((size_t)(b * TDIM + kb0 + half * 2) * KVH + kvh) * DH + l16;
#pragma unroll
    for (int kk = 0; kk < 4; ++kk) {
      v2f pa;                                        // A-layout read of P
      pa.x = pbuf[l16 * 16 + kk * 4 + half * 2 + 0];
      pa.y = pbuf[l16 * 16 + kk * 4 + half * 2 + 1];
      const float* vk = vrow + (size_t)(kk * 4) * (KVH * DH);
      { v2f vb; vb.x = vk[0];  vb.y = vk[KVH * DH + 0];  o0 = wmma4(pa, vb, o0); }
      { v2f vb; vb.x = vk[16]; vb.y = vk[KVH * DH + 16]; o1 = wmma4(pa, vb, o1); }
      { v2f vb; vb.x = vk[32]; vb.y = vk[KVH * DH + 32]; o2 = wmma4(pa, vb, o2); }
      { v2f vb; vb.x = vk[48]; vb.y = vk[KVH * DH + 48]; o3 = wmma4(pa, vb, o3); }
    }
  }

  // Normalize and store [B,T,H,D].
  float* op = O + ((size_t)(b * TDIM + qbase + half * 8) * HH + h) * DH + l16;
#pragma unroll
  for (int r = 0; r < 8; ++r) {
    float il = 1.0f / l[r];
    size_t ro = (size_t)r * HH * DH;
    op[ro +  0] = o0[r] * il;
    op[ro + 16] = o1[r] * il;
    op[ro + 32] = o2[r] * il;
    op[ro + 48] = o3[r] * il;
  }
}

// ---------------------------------------------------------------------------
extern "C" void kernel_launch(void* const* d_in, const int* in_sizes, int n_in,
                              void* d_out, int out_size, void* d_ws, size_t ws_size,
                              hipStream_t stream) {
  const float* x  = (const float*)d_in[0];
  const float* fc = (const float*)d_in[1];
  const float* Wq = (const float*)d_in[2];
  const float* Wk = (const float*)d_in[3];
  const float* Wv = (const float*)d_in[4];
  const float* Wo = (const float*)d_in[5];
  float* out = (float*)d_out;

  const size_t BT = (size_t)BDIM * TDIM;            // 4096
  float* q  = (float*)d_ws;                         // [BT, H*D]
  float* k  = q  + BT * HH  * DH;                   // [BT, KV*D]
  float* v  = k  + BT * KVH * DH;                   // [BT, KV*D]
  float* ao = v  + BT * KVH * DH;                   // [BT, H*D]

  auto gblocks = [](int M, int N) {
    int tiles = (M / 16) * (N / (16 * CT));
    return (tiles + 7) / 8;                         // 8 waves per 256-thread block
  };

  // QKV projections
  gemm16x128_f32<<<gblocks((int)BT, HH * DH),  256, 0, stream>>>(x, Wq, q, (int)BT, EDIM, HH * DH);
  gemm16x128_f32<<<gblocks((int)BT, KVH * DH), 256, 0, stream>>>(x, Wk, k, (int)BT, EDIM, KVH * DH);
  gemm16x128_f32<<<gblocks((int)BT, KVH * DH), 256, 0, stream>>>(x, Wv, v, (int)BT, EDIM, KVH * DH);

  // RoPE on q and k
  int qpairs = (int)(BT * HH  * DH / 2);
  int kpairs = (int)(BT * KVH * DH / 2);
  rope_apply<<<(qpairs + 255) / 256, 256, 0, stream>>>(q, fc, HH,  qpairs);
  rope_apply<<<(kpairs + 255) / 256, 256, 0, stream>>>(k, fc, KVH, kpairs);

  // Causal GQA attention: one wave per (b, h, 16-query tile)
  gqa_attention<<<BDIM * HH * (TDIM / 16), 32, 0, stream>>>(q, k, v, ao);

  // Output projection
  gemm16x128_f32<<<gblocks((int)BT, EDIM), 256, 0, stream>>>(ao, Wo, out, (int)BT, HH * DH, EDIM);
}